// OrdinalRegressionTuned_63230508532388
// MI455X (gfx1250) — compile-verified
//
#include <hip/hip_runtime.h>

typedef __attribute__((ext_vector_type(2))) float v2f;
typedef __attribute__((ext_vector_type(4))) float v4f;
typedef __attribute__((ext_vector_type(8))) float v8f;
typedef __attribute__((ext_vector_type(2))) int   v2i;

#ifndef NUM_CLASSES
#define NUM_CLASSES 6
#endif

// Kernel 1: each thread processes 2 rows (48 B of predictions, 8 B of targets),
// accumulates sum of squared (pred*weight - onehot) over both rows, then the
// wave reduces 32 lane-accumulators with one V_WMMA_F32_16X16X4_F32 and one
// shfl_xor(16). Per-block partial goes to d_ws.
__global__ __launch_bounds__(256) void ord_partial_kernel(
    const float* __restrict__ pred,
    const int*   __restrict__ tgt,
    float*       __restrict__ partial,
    int nPairs)
{
    const int gid = blockIdx.x * blockDim.x + threadIdx.x;
    float acc = 0.0f;

    if (gid < nPairs) {
        // 2 rows => 12 floats => 3 x b128, 16B-aligned (48*gid % 16 == 0)
        const v4f* p4 = reinterpret_cast<const v4f*>(pred) + (size_t)gid * 3;
        v4f va = __builtin_nontemporal_load(p4 + 0);
        v4f vb = __builtin_nontemporal_load(p4 + 1);
        v4f vc = __builtin_nontemporal_load(p4 + 2);
        const v2i* t2 = reinterpret_cast<const v2i*>(tgt) + gid;
        v2i t = __builtin_nontemporal_load(t2);

        const float p0[NUM_CLASSES] = { va[0], va[1], va[2], va[3], vb[0], vb[1] };
        const float p1[NUM_CLASSES] = { vb[2], vb[3], vc[0], vc[1], vc[2], vc[3] };
        const int   t0i = t[0];
        const int   t1i = t[1];
        const float t0f = (float)t0i;
        const float t1f = (float)t1i;

        #pragma unroll
        for (int j = 0; j < NUM_CLASSES; ++j) {
            const float jf = (float)j;
            float w0 = fabsf(t0f + 1.0f - jf);
            float d0 = p0[j] * w0 - ((j == t0i) ? 1.0f : 0.0f);
            acc = fmaf(d0, d0, acc);
            float w1 = fabsf(t1f + 1.0f - jf);
            float d1 = p1[j] * w1 - ((j == t1i) ? 1.0f : 0.0f);
            acc = fmaf(d1, d1, acc);
        }
    }

    // ---- wave32 reduction via WMMA: D[m][n] = acc_m + acc_{m+16} ----
    // A (16x4 f32, 2 VGPRs/lane): lane L<16 holds {A[L][0],A[L][1]},
    // lane L>=16 holds {A[L-16][2],A[L-16][3]}. Put acc in slot 0, zero slot 1.
    v2f A; A[0] = acc;  A[1] = 0.0f;
    v2f B; B[0] = 1.0f; B[1] = 1.0f;   // all-ones 4x16
    v8f C = {};
    v8f D = __builtin_amdgcn_wmma_f32_16x16x4_f32(
        /*neg_a=*/false, A, /*neg_b=*/false, B,
        /*c_mod=*/(short)0, C, /*reuse_a=*/false, /*reuse_b=*/false);

    // Lane n (<16) holds D[0..7][n] = sum_{m=0..7}(acc_m+acc_{m+16}) per VGPR;
    // lane n+16 holds D[8..15][n]. Sum 8 VGPRs then fold halves.
    float s = ((D[0] + D[1]) + (D[2] + D[3])) + ((D[4] + D[5]) + (D[6] + D[7]));
    s += __shfl_xor(s, 16, 32);        // every lane now has the wave-32 total

    __shared__ float lds[8];
    const int lane = threadIdx.x & 31;
    const int wave = threadIdx.x >> 5;
    if (lane == 0) lds[wave] = s;
    __syncthreads();
    if (threadIdx.x == 0) {
        float blk = 0.0f;
        #pragma unroll
        for (int w = 0; w < 8; ++w) blk += lds[w];
        partial[blockIdx.x] = blk;
    }
}

// Kernel 2: single block folds the per-block partials and applies 1/N.
__global__ __launch_bounds__(256) void ord_final_kernel(
    const float* __restrict__ partial,
    int nPart,
    float* __restrict__ out,
    float invN)
{
    float s = 0.0f;
    for (int i = threadIdx.x; i < nPart; i += 256) s += partial[i];

    #pragma unroll
    for (int m = 16; m >= 1; m >>= 1) s += __shfl_xor(s, m, 32);

    __shared__ float lds[8];
    if ((threadIdx.x & 31) == 0) lds[threadIdx.x >> 5] = s;
    __syncthreads();
    if (threadIdx.x == 0) {
        float tot = 0.0f;
        #pragma unroll
        for (int w = 0; w < 8; ++w) tot += lds[w];
        out[0] = tot * invN;
    }
}

extern "C" void kernel_launch(void* const* d_in, const int* in_sizes, int n_in,
                              void* d_out, int out_size, void* d_ws, size_t ws_size,
                              hipStream_t stream) {
    const float* pred = (const float*)d_in[0];   // [N,6] float32
    const int*   tgt  = (const int*)d_in[1];     // [N] integer targets
    const int N      = in_sizes[1];              // 4,194,304
    const int nPairs = N / 2;                    // 2 rows per thread
    const int threads = 256;
    const int blocks  = (nPairs + threads - 1) / threads;  // 8192

    float* partials = (float*)d_ws;              // 8192 floats = 32 KB scratch

    ord_partial_kernel<<<blocks, threads, 0, stream>>>(pred, tgt, partials, nPairs);
    ord_final_kernel<<<1, 256, 0, stream>>>(partials, blocks, (float*)d_out,
                                            1.0f / (float)N);
}